// StructuredMaskedCE_27616639713924
// MI455X (gfx1250) — compile-verified
//
#include <hip/hip_runtime.h>
#include <hip/hip_bf16.h>
#include <math.h>

// ---- problem constants (match the reference) ----
#define NB 8                 // proteins per batch
#define LRES 600             // residues per protein
#define NA (3 * LRES)        // atoms per segment = 1800
#define NPAIR ((NA * (NA - 1)) / 2)   // 1,619,100 pairs per segment

#define NT 113                         // ceil(NA / 16) tiles per dimension
#define NTILES ((NT * (NT + 1)) / 2)   // 6441 upper-triangle tiles per segment
#define WAVES_PER_BLOCK 8

typedef __attribute__((ext_vector_type(2))) float v2f;
typedef __attribute__((ext_vector_type(8))) float v8f;

// ---------------------------------------------------------------------------
// zero the per-segment error accumulators
// ---------------------------------------------------------------------------
__global__ void smce_init_ws(float* __restrict__ ws) {
    int t = threadIdx.x;
    if (t < NB) ws[t] = 0.0f;
}

// ---------------------------------------------------------------------------
// main kernel: one 16x16 (i,j) tile of the pair-distance matrix per wave.
// Gram trick: d^2 = |ci|^2 + |cj|^2 - 2 * (ci . cj); the 16x16 block of
// inner products is one V_WMMA_F32_16X16X4_F32 (K=4, 4th coord zero-padded).
// Fully branchless inner loop; no LDS, no block barrier.
// ---------------------------------------------------------------------------
__global__ __launch_bounds__(256) void smce_pair_rmsd_tiles(
    const float* __restrict__ inputs,   // (NB * NPAIR)
    const float* __restrict__ target,   // (NB * NA, 3)
    const float* __restrict__ mask,     // (NB * NA)
    float* __restrict__ ws)             // (NB) partial sums of err
{
    const int lane = threadIdx.x & 31;
    const int w    = threadIdx.x >> 5;
    const int b    = blockIdx.y;

    const int  t         = blockIdx.x * WAVES_PER_BLOCK + w;
    const bool validTile = (t < NTILES);
    const int  tt        = validTile ? t : 0;

    // invert triangular tile index: rowstart(ti) = ti*NT - ti*(ti-1)/2
    int ti = (int)((2.0f * NT + 1.0f -
                    __builtin_amdgcn_sqrtf((2.0f * NT + 1.0f) * (2.0f * NT + 1.0f)
                                           - 8.0f * (float)tt)) * 0.5f);
    if (ti < 0) ti = 0;
    if (ti > NT - 1) ti = NT - 1;
    while (ti + 1 < NT && ((ti + 1) * NT - (((ti + 1) * ti) >> 1)) <= tt) ++ti;
    while (ti > 0 && (ti * NT - ((ti * (ti - 1)) >> 1)) > tt) --ti;
    const int tj = ti + (tt - (ti * NT - ((ti * (ti - 1)) >> 1)));

    // per-lane row/col atom indices for this tile
    const int rI = ti * 16 + (lane & 15);
    const int rJ = tj * 16 + (lane & 15);

    const float* cb    = target + (size_t)b * NA * 3;
    const float* mb    = mask   + (size_t)b * NA;
    const float* predb = inputs + (size_t)b * NPAIR;

    // clamp OOB rows to 0 and zero their mask -> contributions vanish via pm
    const int  rIc = (rI < NA) ? rI : 0;
    const int  rJc = (rJ < NA) ? rJ : 0;
    const float cix = cb[rIc * 3 + 0], ciy = cb[rIc * 3 + 1], ciz = cb[rIc * 3 + 2];
    const float cjx = cb[rJc * 3 + 0], cjy = cb[rJc * 3 + 1], cjz = cb[rJc * 3 + 2];
    const float miL = (rI < NA) ? mb[rIc] : 0.0f;   // this lane's row mask
    const float mj  = (rJ < NA) ? mb[rJc] : 0.0f;

    const float ri2L = cix * cix + ciy * ciy + ciz * ciz;  // this lane's row |c|^2
    const float rj2  = cjx * cjx + cjy * cjy + cjz * cjz;

    const int mbase = (lane >= 16) ? 8 : 0;

    // prefetch the predicted-distance row this lane will read first
    {
        int i0 = ti * 16 + mbase;
        int j0 = rJ;
        if (i0 < j0 && j0 < NA) {
            int p0 = i0 * (NA - 1) - ((i0 * (i0 - 1)) >> 1) + (j0 - i0 - 1);
            __builtin_prefetch(predb + p0, 0, 1);   // -> global_prefetch_b8
        }
    }

    // A-matrix 16x4 f32 fragment (ISA 7.12.2): VGPR0 = K0 (lanes 0-15) / K2
    // (lanes 16-31); VGPR1 = K1 / K3. K3 is the zero pad.
    v2f afrag, bfrag;
    afrag.x = (lane < 16) ? cix : ciz;
    afrag.y = (lane < 16) ? ciy : 0.0f;
    // B-matrix 4x16 f32 fragment: VGPR0 = row K0 (lanes 0-15) / K1 (16-31);
    // VGPR1 = K2 / K3(=0). N = lane % 16.
    bfrag.x = (lane < 16) ? cjx : cjy;
    bfrag.y = (lane < 16) ? cjz : 0.0f;

    v8f c = {};
    // D = A(16x4) x B(4x16) + 0  -> 16x16 block of ci . cj
    c = __builtin_amdgcn_wmma_f32_16x16x4_f32(
            /*neg_a=*/false, afrag, /*neg_b=*/false, bfrag,
            /*c_mod=*/(short)0, c, /*reuse_a=*/false, /*reuse_b=*/false);

    // C/D layout: lane l, VGPR v -> N = l%16, M = v + 8*(l>=16).
    // Row-side values for local row li were computed by lane li (0..15):
    // fetch via wave shuffle instead of LDS.
    float acc = 0.0f;
#pragma unroll
    for (int v = 0; v < 8; ++v) {
        const int   li   = v + mbase;           // local row 0..15
        const int   i    = ti * 16 + li;
        const int   j    = rJ;
        const bool  ok   = validTile & (i < j) & (j < NA);
        const float ri2  = __shfl(ri2L, li, 32);
        const float mi   = __shfl(miL,  li, 32);
        const float g    = c[v];
        const float d2   = ri2 + rj2 - 2.0f * g;
        const float d    = __builtin_amdgcn_sqrtf(fmaxf(d2, 0.0f));
        const float pm   = mi * mj;
        int p = i * (NA - 1) - ((i * (i - 1)) >> 1) + (j - i - 1);
        p = ok ? p : 0;                         // clamp -> load always in-bounds
        const float pred = predb[p];
        float e = pm * (pred - d);              // (pm*pred - pm*d)
        e = ok ? e : 0.0f;
        acc += e * e;
    }

    // wave32 reduction
#pragma unroll
    for (int off = 16; off > 0; off >>= 1)
        acc += __shfl_xor(acc, off, 32);

    if (lane == 0 && validTile)
        atomicAdd(&ws[b], acc);
}

// ---------------------------------------------------------------------------
// finalize: mean_b sqrt(mean_p err + 1e-6)
// ---------------------------------------------------------------------------
__global__ void smce_finalize(const float* __restrict__ ws, float* __restrict__ out) {
    if (threadIdx.x == 0 && blockIdx.x == 0) {
        float s = 0.0f;
        for (int b = 0; b < NB; ++b)
            s += sqrtf(ws[b] * (1.0f / (float)NPAIR) + 1e-6f);
        out[0] = s * (1.0f / (float)NB);
    }
}

// ---------------------------------------------------------------------------
extern "C" void kernel_launch(void* const* d_in, const int* in_sizes, int n_in,
                              void* d_out, int out_size, void* d_ws, size_t ws_size,
                              hipStream_t stream) {
    const float* inputs = (const float*)d_in[0];   // (NB*NPAIR,) f32
    const float* target = (const float*)d_in[1];   // (NB*NA, 3)  f32
    const float* mask   = (const float*)d_in[2];   // (NB*NA, 1)  f32
    // d_in[3] = indices (int32): fixed equal-sized segments -> not needed

    float* ws  = (float*)d_ws;    // NB partial sums
    float* out = (float*)d_out;   // scalar

    smce_init_ws<<<1, 32, 0, stream>>>(ws);

    dim3 grid((NTILES + WAVES_PER_BLOCK - 1) / WAVES_PER_BLOCK, NB);
    smce_pair_rmsd_tiles<<<grid, 32 * WAVES_PER_BLOCK, 0, stream>>>(inputs, target, mask, ws);

    smce_finalize<<<1, 32, 0, stream>>>(ws, out);
}